// Net_86749749445069
// MI455X (gfx1250) — compile-verified
//
#include <hip/hip_runtime.h>

typedef __attribute__((ext_vector_type(2))) float v2f;
typedef __attribute__((ext_vector_type(8))) float v8f;

#define N_NODES 100000
#define N_EDGES 1600000
#define K1PAD 12   // layer-1 K padded 10 -> 12 (multiple of 4)

__device__ __forceinline__ void atomAddF32(float* p, float v) {
    // non-returning relaxed agent-scope fadd -> global_atomic_add_f32
    __hip_atomic_fetch_add(p, v, __ATOMIC_RELAXED, __HIP_MEMORY_SCOPE_AGENT);
}

// ---------------- dual GEMM via f32 WMMA ----------------
// H = X @ W            [N_NODES, NOUT]   (message source)
// A = X @ V + b        [N_NODES, NOUT]   (aggregation init)
// X: [N_NODES, K] row-major (K % 4 == 0), W/V: [K, NOUT] row-major.
// W/V are staged transposed+padded into LDS once per block:
//   sWt[col*K + k] = W[k*NOUT + col]  (zeros for col >= NOUT)
// so a lane's (k,k+1) B-fragment pair is one ds_load_b64.
// One wave per 16-row tile; v_wmma_f32_16x16x4_f32 over K in steps of 4.
template <int K, int NOUT>
__global__ void gemm_dual_kernel(const float* __restrict__ X,
                                 const float* __restrict__ W,
                                 const float* __restrict__ V,
                                 const float* __restrict__ b,
                                 float* __restrict__ H,
                                 float* __restrict__ A) {
    constexpr int NT    = (NOUT + 15) / 16;
    constexpr int BCOLS = NT * 16;
    __shared__ float sWt[BCOLS * K];
    __shared__ float sVt[BCOLS * K];

    // stage B operands transposed into LDS (8-way reuse across waves)
    for (int idx = threadIdx.x; idx < BCOLS * K; idx += blockDim.x) {
        const int col = idx / K;
        const int k   = idx % K;
        float w = 0.f, v = 0.f;
        if (col < NOUT) {
            w = W[k * NOUT + col];
            v = V[k * NOUT + col];
        }
        sWt[idx] = w;
        sVt[idx] = v;
    }
    __syncthreads();

    const int nTiles = N_NODES / 16;  // 6250, exact
    const int wave = blockIdx.x * (blockDim.x >> 5) + (threadIdx.x >> 5);
    if (wave >= nTiles) return;  // wave-uniform guard, EXEC stays full

    const int lane  = threadIdx.x & 31;
    const int mrow  = lane & 15;        // row within tile (A), col within tile (B)
    const int khalf = (lane >> 4) * 2;  // lanes 16..31 hold K=+2,+3
    const int rowBase = wave * 16;
    const float* xrow = X + (size_t)(rowBase + mrow) * K;

    v8f accW[NT];
    v8f accV[NT];
#pragma unroll
    for (int t = 0; t < NT; ++t) {
        accW[t] = (v8f){0.f, 0.f, 0.f, 0.f, 0.f, 0.f, 0.f, 0.f};
        accV[t] = (v8f){0.f, 0.f, 0.f, 0.f, 0.f, 0.f, 0.f, 0.f};
    }

#pragma unroll
    for (int k0 = 0; k0 < K; k0 += 4) {
        const int kk = k0 + khalf;
        // A fragment: 16x4 f32, lane holds (row=mrow, K=kk, kk+1), 8B aligned
        const v2f a = *(const v2f*)(xrow + kk);
#pragma unroll
        for (int t = 0; t < NT; ++t) {
            const int boff = (t * 16 + mrow) * K + kk;  // contiguous (k,k+1)
            const v2f bw = *(const v2f*)(sWt + boff);   // ds_load_b64
            const v2f bv = *(const v2f*)(sVt + boff);
            accW[t] = __builtin_amdgcn_wmma_f32_16x16x4_f32(
                false, a, false, bw, (short)0, accW[t], false, false);
            accV[t] = __builtin_amdgcn_wmma_f32_16x16x4_f32(
                false, a, false, bv, (short)0, accV[t], false, false);
        }
    }

    // C/D layout: VGPR i holds M = i + 8*(lane>=16), N = lane&15
    const int mBase = (lane >> 4) * 8;
#pragma unroll
    for (int t = 0; t < NT; ++t) {
        const int col = t * 16 + mrow;
        if (col < NOUT) {
            const float bb = b[col];
#pragma unroll
            for (int i = 0; i < 8; ++i) {
                const int r = rowBase + mBase + i;
                H[(size_t)r * NOUT + col] = accW[t][i];
                A[(size_t)r * NOUT + col] = accV[t][i] + bb;
            }
        }
    }
}

// ---------------- input padding: x [N,10] -> xp [N,12] (zero-padded) ----------------
__global__ void padx_kernel(const float* __restrict__ x, float* __restrict__ xp) {
    int i = blockIdx.x * blockDim.x + threadIdx.x;
    if (i < N_NODES * K1PAD) {
        int row = i / K1PAD;
        int c   = i % K1PAD;
        xp[i] = (c < 10) ? x[row * 10 + c] : 0.f;
    }
}

// ---------------- graph-norm kernels ----------------
__global__ void zero_kernel(float* __restrict__ p, int n) {
    int i = blockIdx.x * blockDim.x + threadIdx.x;
    if (i < n) p[i] = 0.f;
}

__global__ void deg_kernel(const long long* __restrict__ ei, float* __restrict__ deg) {
    int e = blockIdx.x * blockDim.x + threadIdx.x;
    if (e < N_EDGES) {
        int d = (int)ei[N_EDGES + e];  // dst row
        atomAddF32(&deg[d], 1.0f);
    }
}

__global__ void dinv_kernel(float* __restrict__ deg) {
    int n = blockIdx.x * blockDim.x + threadIdx.x;
    if (n < N_NODES) {
        float v = deg[n];
        deg[n] = (v > 0.f) ? rsqrtf(v) : 0.f;
    }
}

// ---------------- edge scatter: A[dst] += norm * H[src] ----------------
// 64-wide features: 16 lanes per edge, float4 per lane.
__global__ void scatter64_kernel(const long long* __restrict__ ei,
                                 const float* __restrict__ dinv,
                                 const float* __restrict__ H,
                                 float* __restrict__ A) {
    long long tid = (long long)blockIdx.x * blockDim.x + threadIdx.x;
    int e  = (int)(tid >> 4);
    int f0 = (int)(tid & 15) << 2;
    if (e < N_EDGES) {
        int s = (int)ei[e];
        int d = (int)ei[N_EDGES + e];
        float nrm = dinv[s] * dinv[d];
        float4 h = *(const float4*)(H + (size_t)s * 64 + f0);
        float* ap = A + (size_t)d * 64 + f0;
        atomAddF32(ap + 0, nrm * h.x);
        atomAddF32(ap + 1, nrm * h.y);
        atomAddF32(ap + 2, nrm * h.z);
        atomAddF32(ap + 3, nrm * h.w);
    }
}

// 2-wide features: one lane per edge.
__global__ void scatter2_kernel(const long long* __restrict__ ei,
                                const float* __restrict__ dinv,
                                const float* __restrict__ H,
                                float* __restrict__ A) {
    int e = blockIdx.x * blockDim.x + threadIdx.x;
    if (e < N_EDGES) {
        int s = (int)ei[e];
        int d = (int)ei[N_EDGES + e];
        float nrm = dinv[s] * dinv[d];
        atomAddF32(&A[d * 2 + 0], nrm * H[s * 2 + 0]);
        atomAddF32(&A[d * 2 + 1], nrm * H[s * 2 + 1]);
    }
}

__global__ void relu_kernel(float* __restrict__ p, int n) {
    int i = blockIdx.x * blockDim.x + threadIdx.x;
    if (i < n) p[i] = fmaxf(p[i], 0.f);
}

__global__ void logsoftmax2_kernel(const float* __restrict__ A3, float* __restrict__ out) {
    int n = blockIdx.x * blockDim.x + threadIdx.x;
    if (n < N_NODES) {
        float o0 = A3[n * 2 + 0];
        float o1 = A3[n * 2 + 1];
        float m = fmaxf(o0, o1);
        float lse = m + logf(expf(o0 - m) + expf(o1 - m));
        out[n * 2 + 0] = o0 - lse;
        out[n * 2 + 1] = o1 - lse;
    }
}

extern "C" void kernel_launch(void* const* d_in, const int* in_sizes, int n_in,
                              void* d_out, int out_size, void* d_ws, size_t ws_size,
                              hipStream_t stream) {
    const float*     x  = (const float*)d_in[0];
    const long long* ei = (const long long*)d_in[1];  // int64 [2, E]
    const float* W1 = (const float*)d_in[2];
    const float* V1 = (const float*)d_in[3];
    const float* b1 = (const float*)d_in[4];
    const float* W2 = (const float*)d_in[5];
    const float* V2 = (const float*)d_in[6];
    const float* b2 = (const float*)d_in[7];
    const float* W3 = (const float*)d_in[8];
    const float* V3 = (const float*)d_in[9];
    const float* b3 = (const float*)d_in[10];
    float* out = (float*)d_out;

    // workspace layout (floats), 64-element aligned chunks
    float* ws = (float*)d_ws;
    size_t off = 0;
    float* dinv = ws + off; off += ((size_t)N_NODES + 63) & ~(size_t)63;
    float* xp   = ws + off; off += (size_t)N_NODES * K1PAD;  // padded input
    float* h    = ws + off; off += (size_t)N_NODES * 64;     // shared h1/h2 buffer
    float* a1   = ws + off; off += (size_t)N_NODES * 64;
    float* a2   = ws + off; off += (size_t)N_NODES * 64;
    float* h3   = ws + off; off += (size_t)N_NODES * 2;
    float* a3   = ws + off; off += (size_t)N_NODES * 2;
    (void)ws_size; (void)in_sizes; (void)n_in; (void)out_size;

    const int nTiles = N_NODES / 16;                 // 6250
    const int gemmBlocks = (nTiles + 7) / 8;         // 8 waves / 256-thread block
    const int nodeBlocks = (N_NODES + 255) / 256;
    const int edgeBlocks = (N_EDGES + 255) / 256;
    const int feat64Blk  = (int)(((long long)N_EDGES * 16 + 255) / 256);  // 100000
    const int reluBlocks = (N_NODES * 64 + 255) / 256;
    const int padBlocks  = (N_NODES * K1PAD + 255) / 256;

    // gcn_norm: deg -> dinv (in place); pad x in parallel stream order
    zero_kernel<<<nodeBlocks, 256, 0, stream>>>(dinv, N_NODES);
    padx_kernel<<<padBlocks, 256, 0, stream>>>(x, xp);
    deg_kernel<<<edgeBlocks, 256, 0, stream>>>(ei, dinv);
    dinv_kernel<<<nodeBlocks, 256, 0, stream>>>(dinv);

    // layer 1: 10(->12 padded) -> 64
    gemm_dual_kernel<K1PAD, 64><<<gemmBlocks, 256, 0, stream>>>(xp, W1, V1, b1, h, a1);
    scatter64_kernel<<<feat64Blk, 256, 0, stream>>>(ei, dinv, h, a1);
    relu_kernel<<<reluBlocks, 256, 0, stream>>>(a1, N_NODES * 64);

    // layer 2: 64 -> 64 (reuse h buffer)
    gemm_dual_kernel<64, 64><<<gemmBlocks, 256, 0, stream>>>(a1, W2, V2, b2, h, a2);
    scatter64_kernel<<<feat64Blk, 256, 0, stream>>>(ei, dinv, h, a2);
    relu_kernel<<<reluBlocks, 256, 0, stream>>>(a2, N_NODES * 64);

    // layer 3: 64 -> 2, then log_softmax
    gemm_dual_kernel<64, 2><<<gemmBlocks, 256, 0, stream>>>(a2, W3, V3, b3, h3, a3);
    scatter2_kernel<<<edgeBlocks, 256, 0, stream>>>(ei, dinv, h3, a3);
    logsoftmax2_kernel<<<nodeBlocks, 256, 0, stream>>>(a3, out);
}